// ChebConv_67594195304931
// MI455X (gfx1250) — compile-verified
//
#include <hip/hip_runtime.h>
#include <hip/hip_bf16.h>

// ---------------------------------------------------------------------------
// ChebConv (K=4) for MI455X / gfx1250, wave32.
//   X0 = transpose(x)            (V, B*Cin) = (V,512)
//   X1 = S X0 ; X2 = 2 S X1 - X0 ; X3 = 2 S X2 - X1   (SpMM via f32 atomics)
//   out[b,o,v] = bias[o] + sum_k (Xk[v*4+b, :] @ Wk[:, o])   via WMMA f32 16x16x4
// ---------------------------------------------------------------------------

typedef float v2f __attribute__((ext_vector_type(2)));
typedef float v8f __attribute__((ext_vector_type(8)));

#define CIN  128
#define COUT 128
#define BSZ  4
#define FW   512   // B*Cin feature width

// ---- 1) transpose (512 rows x V cols) -> (V rows x 512 cols), LDS tiled ----
__global__ void cheb_transpose(const float* __restrict__ in,   // (512, V)
                               float* __restrict__ out,        // (V, 512)
                               int Vv) {
    __shared__ float tile[32][33];
    const int vt = blockIdx.x * 32;     // v tile
    const int rt = blockIdx.y * 32;     // bc tile (512/32 = 16 blocks, always full)
    const int tx = threadIdx.x;         // 0..31
    const int ty = threadIdx.y;         // 0..7
#pragma unroll
    for (int i = 0; i < 32; i += 8) {
        const int v = vt + tx;
        const int r = rt + ty + i;
        if (v < Vv) tile[ty + i][tx] = in[(size_t)r * Vv + v];
    }
    __syncthreads();
#pragma unroll
    for (int i = 0; i < 32; i += 8) {
        const int v = vt + ty + i;
        const int r = rt + tx;
        if (v < Vv) out[(size_t)v * FW + r] = tile[tx][ty + i];
    }
}

// ---- 2a) y = beta * xprev (beta = 0 or -1), float4 vectorized ----
__global__ void cheb_scale_init(float* __restrict__ y,
                                const float* __restrict__ xprev,
                                float beta, int n4) {
    const int i = blockIdx.x * blockDim.x + threadIdx.x;
    if (i >= n4) return;
    const float4 p = reinterpret_cast<const float4*>(xprev)[i];
    float4 r;
    r.x = beta * p.x; r.y = beta * p.y; r.z = beta * p.z; r.w = beta * p.w;
    reinterpret_cast<float4*>(y)[i] = r;
}

// ---- 2b) y[row,:] += alpha * val * x[col,:] per edge (32 lanes / edge) ----
__global__ void cheb_edge_scatter(float* __restrict__ y,
                                  const float* __restrict__ x,
                                  const int* __restrict__ rows,
                                  const int* __restrict__ cols,
                                  const float* __restrict__ vals,
                                  float alpha, int E) {
    const int lane = threadIdx.x & 31;
    const int e = blockIdx.x * (blockDim.x >> 5) + (threadIdx.x >> 5);
    if (e >= E) return;
    const int   r  = rows[e];
    const int   c  = cols[e];
    const float av = alpha * vals[e];
    const float* __restrict__ xs = x + (size_t)c * FW;
    float* __restrict__ yd = y + (size_t)r * FW;
#pragma unroll 4
    for (int f = lane; f < FW; f += 32) {
        __hip_atomic_fetch_add(&yd[f], av * xs[f],
                               __ATOMIC_RELAXED, __HIP_MEMORY_SCOPE_AGENT);
    }
}

// ---- 3) GEMM: C(M,128) = [X0|X1|X2|X3](M,512) @ W(512,128), fused bias +
//         scattered store to out(B, Cout, V).  One wave = 16 rows x 128 cols.
__global__ void cheb_wmma_gemm(const float* __restrict__ X,   // 4 * (M,128)
                               const float* __restrict__ W,   // (4,128,128)
                               const float* __restrict__ bias,// (128,)
                               float* __restrict__ out,       // (4,128,V)
                               int M, int Vv) {
    const int lane  = threadIdx.x & 31;
    const int wave  = threadIdx.x >> 5;
    const int m0    = (blockIdx.x * 8 + wave) * 16;
    if (m0 >= M) return;

    const int mrow  = lane & 15;        // A: M index, C: N index
    const int khalf = (lane >> 4) * 2;  // A/B: K pair selector (0 or 2)

    v8f acc[8] = {};                    // 8 n-tiles of 16 -> full 128 columns

    for (int k = 0; k < 4; ++k) {
        const float* __restrict__ A  = X + (size_t)k * M * CIN;
        const float* __restrict__ Wk = W + k * CIN * COUT;
        const float* __restrict__ Arow = A + (size_t)(m0 + mrow) * CIN;
#pragma unroll 1
        for (int kk = 0; kk < CIN; kk += 4) {
            // A fragment: 16x4, lane = M, (lane>>4)*2 + j = K  (ISA 7.12.2)
            const v2f a = *reinterpret_cast<const v2f*>(Arow + kk + khalf);
            const float* __restrict__ wr0 = Wk + (size_t)(kk + khalf) * COUT;
#pragma unroll
            for (int nt = 0; nt < 8; ++nt) {
                // B fragment: 4x16, row K striped across lanes (n = lane&15)
                v2f b;
                b.x = wr0[nt * 16 + mrow];
                b.y = wr0[COUT + nt * 16 + mrow];
                acc[nt] = __builtin_amdgcn_wmma_f32_16x16x4_f32(
                    false, a, false, b, (short)0, acc[nt], false, false);
            }
        }
    }

    // C/D layout: VGPR r, lanes 0-15 -> M=r, lanes 16-31 -> M=8+r; N = lane&15
    const int mhi = (lane >> 4) * 8;
#pragma unroll
    for (int nt = 0; nt < 8; ++nt) {
        const int o = nt * 16 + mrow;
        const float bo = bias[o];
#pragma unroll
        for (int r = 0; r < 8; ++r) {
            const int m  = m0 + mhi + r;
            const int v  = m >> 2;      // m = v*4 + b
            const int bb = m & 3;
            out[((size_t)bb * COUT + o) * Vv + v] = acc[nt][r] + bo;
        }
    }
}

// ---------------------------------------------------------------------------
extern "C" void kernel_launch(void* const* d_in, const int* in_sizes, int n_in,
                              void* d_out, int out_size, void* d_ws, size_t ws_size,
                              hipStream_t stream) {
    const float* x     = (const float*)d_in[0];   // (B,Cin,V) f32
    const int*   erow  = (const int*)  d_in[1];   // (E,)
    const int*   ecol  = (const int*)  d_in[2];   // (E,)
    const float* evals = (const float*)d_in[3];   // (E,)
    const float* w     = (const float*)d_in[4];   // (K,Cin,Cout)
    const float* bias  = (const float*)d_in[5];   // (Cout,)
    float*       out   = (float*)d_out;           // (B,Cout,V)

    const int Vv = in_sizes[0] / FW;              // B*Cin*V / 512
    const int E  = in_sizes[1];
    const int M  = Vv * BSZ;                      // 200000

    const size_t bufElems = (size_t)Vv * FW;      // 25.6M floats per buffer
    float* X0 = (float*)d_ws;
    float* X1 = X0 + bufElems;
    float* X2 = X1 + bufElems;
    float* X3 = X2 + bufElems;

    // X0 = transpose(x)
    cheb_transpose<<<dim3((Vv + 31) / 32, FW / 32), dim3(32, 8), 0, stream>>>(
        x, X0, Vv);

    const int n4 = (int)(bufElems / 4);
    const dim3 initGrid((n4 + 255) / 256);
    const dim3 edgeGrid((E + 7) / 8);             // 8 edges per 256-thread block

    // X1 = S X0
    cheb_scale_init<<<initGrid, 256, 0, stream>>>(X1, X0, 0.0f, n4);
    cheb_edge_scatter<<<edgeGrid, 256, 0, stream>>>(X1, X0, erow, ecol, evals, 1.0f, E);
    // X2 = 2 S X1 - X0
    cheb_scale_init<<<initGrid, 256, 0, stream>>>(X2, X0, -1.0f, n4);
    cheb_edge_scatter<<<edgeGrid, 256, 0, stream>>>(X2, X1, erow, ecol, evals, 2.0f, E);
    // X3 = 2 S X2 - X1
    cheb_scale_init<<<initGrid, 256, 0, stream>>>(X3, X1, -1.0f, n4);
    cheb_edge_scatter<<<edgeGrid, 256, 0, stream>>>(X3, X2, erow, ecol, evals, 2.0f, E);

    // out = [X0|X1|X2|X3] @ W + bias, scattered to (B,Cout,V)
    const int strips = M / 16;                    // 12500
    cheb_wmma_gemm<<<(strips + 7) / 8, 256, 0, stream>>>(X0, w, bias, out, M, Vv);
}